// SpectralConv2d_86225763434528
// MI455X (gfx1250) — compile-verified
//
#include <hip/hip_runtime.h>
#include <math.h>
#include <stdint.h>

typedef __attribute__((ext_vector_type(2))) float v2f;
typedef __attribute__((ext_vector_type(8))) float v8f;

// D(16x16,f32) = A(16x4,f32) x B(4x16,f32) + C
__device__ __forceinline__ v8f wmma_f32(v2f a, v2f b, v8f c) {
    return __builtin_amdgcn_wmma_f32_16x16x4_f32(
        /*neg_a=*/false, a, /*neg_b=*/false, b,
        /*c_mod=*/(short)0, c, /*reuse_a=*/false, /*reuse_b=*/false);
}

// Async global -> LDS copy of one dword (GV mode), tracked by ASYNCcnt.
__device__ __forceinline__ void async_copy_b32(uint32_t lds_off, const float* gptr) {
    asm volatile("global_load_async_to_lds_b32 %0, %1, off"
                 :: "v"(lds_off), "v"(gptr) : "memory");
}
__device__ __forceinline__ void wait_async0() {
    asm volatile("s_wait_asynccnt 0" ::: "memory");
}

// ---------------------------------------------------------------------------
// Basis: CAS[k][n]=cos+sin(2πkn/256), CASm=cos-sin, COS, SIN   (16x256 each)
// ---------------------------------------------------------------------------
__global__ void basis_kernel(float* __restrict__ bas) {
    const int N = 16 * 256;
    int t = blockIdx.x * blockDim.x + threadIdx.x;
    for (int idx = t; idx < N; idx += gridDim.x * blockDim.x) {
        int k = idx >> 8, n = idx & 255;
        float ang = (float)(2.0 * M_PI / 256.0) * (float)((k * n) & 255);
        float s, c;
        sincosf(ang, &s, &c);
        bas[0 * N + idx] = c + s;   // CAS
        bas[1 * N + idx] = c - s;   // CASm
        bas[2 * N + idx] = c;       // COS
        bas[3 * N + idx] = s;       // SIN
    }
}

// ---------------------------------------------------------------------------
// Forward truncated DHT: one block per (b,i) image.
// U = X·CAS^T, V = X·CASm^T (256x256x16), Xh = (COS·U + SIN·V)/256 (16x16)
// ---------------------------------------------------------------------------
__global__ void __launch_bounds__(256) dht_fwd_kernel(const float* __restrict__ x,
                                                      const float* __restrict__ bas,
                                                      float* __restrict__ Xh) {
    __shared__ float casL[16 * 257];   // [k2][n2], padded stride 257
    __shared__ float U[256 * 16];      // [n1][k2]
    __shared__ float V[256 * 16];
    __shared__ float part[8 * 256];    // per-wave partial 16x16 tiles

    const int bi   = blockIdx.x;                 // b*32 + i
    const float* X = x + (size_t)bi * 65536;
    const int tid  = threadIdx.x;
    const int lane = tid & 31, wave = tid >> 5;
    const int half = lane >> 4, lm = lane & 15;

    // Stage CAS basis into LDS via the async DMA path (no VGPR round trip).
    for (int idx = tid; idx < 4096; idx += 256) {
        int k = idx >> 8, n = idx & 255;
        async_copy_b32((uint32_t)(uintptr_t)&casL[k * 257 + n], bas + idx);
    }
    wait_async0();
    __syncthreads();

    // Phase 1: each wave computes M-tiles {wave, wave+8} of U and V.
    for (int t = 0; t < 2; ++t) {
        const int mt  = wave + t * 8;
        const int row = mt * 16 + lm;
        const float* xr = X + row * 256 + 2 * half;   // A frag: K offset per half
        v8f accU = {}; v8f accV = {};
        for (int k = 0; k < 256; k += 4) {
            if ((k & 31) == 0) __builtin_prefetch(xr + k + 64, 0, 0);
            const int kk = k + 2 * half;
            v2f a;  a.x  = xr[k];                     a.y  = xr[k + 1];
            v2f bu; bu.x = casL[lm * 257 + kk];       bu.y = casL[lm * 257 + kk + 1];
            // cas(-θ): negate column index mod 256
            v2f bv; bv.x = casL[lm * 257 + ((256 - kk) & 255)];
                    bv.y = casL[lm * 257 + ((255 - kk) & 255)];
            accU = wmma_f32(a, bu, accU);
            accV = wmma_f32(a, bv, accV);
        }
        for (int r = 0; r < 8; ++r) {
            U[(mt * 16 + r + 8 * half) * 16 + lm] = accU[r];
            V[(mt * 16 + r + 8 * half) * 16 + lm] = accV[r];
        }
    }
    __syncthreads();

    // Phase 2: Xh = COS·U + SIN·V, K=256 split across the 8 waves.
    {
        const float* cosB = bas + 2 * 4096;
        const float* sinB = bas + 3 * 4096;
        v8f acc = {};
        for (int k = wave * 32; k < wave * 32 + 32; k += 4) {
            const int kk = k + 2 * half;
            v2f ac; ac.x = cosB[lm * 256 + kk];  ac.y = cosB[lm * 256 + kk + 1];
            v2f as; as.x = sinB[lm * 256 + kk];  as.y = sinB[lm * 256 + kk + 1];
            v2f bu; bu.x = U[kk * 16 + lm];      bu.y = U[(kk + 1) * 16 + lm];
            v2f bv; bv.x = V[kk * 16 + lm];      bv.y = V[(kk + 1) * 16 + lm];
            acc = wmma_f32(ac, bu, acc);
            acc = wmma_f32(as, bv, acc);
        }
        for (int r = 0; r < 8; ++r)
            part[wave * 256 + (r + 8 * half) * 16 + lm] = acc[r];
    }
    __syncthreads();

    float s = 0.f;
    for (int w = 0; w < 8; ++w) s += part[w * 256 + tid];
    Xh[(size_t)bi * 256 + tid] = s * (1.0f / 256.0f);
}

// ---------------------------------------------------------------------------
// Per-mode channel mix: res_k(16x32) = Xh_k(16x32)·Wp_k + Xh_nk(16x32)·Wm_k
// Wp = 0.5(w+wn), Wm = 0.5(w-wn), wn = w at mode negated mod 16.
// One wave per mode; 256 blocks of 32 threads.
// ---------------------------------------------------------------------------
__global__ void __launch_bounds__(32) mode_mix_kernel(const float* __restrict__ Xh,
                                                      const float* __restrict__ w,
                                                      float* __restrict__ res) {
    const int mode  = blockIdx.x;                 // k1*16 + k2
    const int k1 = mode >> 4, k2 = mode & 15;
    const int nmode = (((16 - k1) & 15) << 4) | ((16 - k2) & 15);
    const int lane = threadIdx.x & 31;
    const int half = lane >> 4, lm = lane & 15;

    v8f acc0 = {}, acc1 = {};
    for (int i = 0; i < 32; i += 4) {
        const int ii = i + 2 * half;
        v2f a;  a.x  = Xh[(lm * 32 + ii) * 256 + mode];
                a.y  = Xh[(lm * 32 + ii + 1) * 256 + mode];
        v2f an; an.x = Xh[(lm * 32 + ii) * 256 + nmode];
                an.y = Xh[(lm * 32 + ii + 1) * 256 + nmode];

        // o-tile 0 (o = lm)
        const float* w0  = w + ((size_t)ii * 32 + lm) * 256;
        const float* w0b = w + ((size_t)(ii + 1) * 32 + lm) * 256;
        v2f bp, bm;
        bp.x = 0.5f * (w0[mode] + w0[nmode]);  bp.y = 0.5f * (w0b[mode] + w0b[nmode]);
        bm.x = 0.5f * (w0[mode] - w0[nmode]);  bm.y = 0.5f * (w0b[mode] - w0b[nmode]);
        acc0 = wmma_f32(a,  bp, acc0);
        acc0 = wmma_f32(an, bm, acc0);

        // o-tile 1 (o = 16 + lm)
        const float* w1  = w + ((size_t)ii * 32 + 16 + lm) * 256;
        const float* w1b = w + ((size_t)(ii + 1) * 32 + 16 + lm) * 256;
        bp.x = 0.5f * (w1[mode] + w1[nmode]);  bp.y = 0.5f * (w1b[mode] + w1b[nmode]);
        bm.x = 0.5f * (w1[mode] - w1[nmode]);  bm.y = 0.5f * (w1b[mode] - w1b[nmode]);
        acc1 = wmma_f32(a,  bp, acc1);
        acc1 = wmma_f32(an, bm, acc1);
    }
    for (int r = 0; r < 8; ++r) {
        const int b = r + 8 * half;
        res[((size_t)b * 32 + lm) * 256 + mode]      = acc0[r];
        res[((size_t)b * 32 + 16 + lm) * 256 + mode] = acc1[r];
    }
}

// ---------------------------------------------------------------------------
// Inverse DHT expansion: one block per (b,o) image.
// P = res·CAS, Q = res·CASm (16x16 -> 16x256), out = (COS^T·P + SIN^T·Q)/256.
// ---------------------------------------------------------------------------
__global__ void __launch_bounds__(256) dht_inv_kernel(const float* __restrict__ res,
                                                      const float* __restrict__ bas,
                                                      float* __restrict__ out) {
    __shared__ float P[16 * 256];   // [k1][n2]
    __shared__ float Q[16 * 256];

    const int bo   = blockIdx.x;                 // b*32 + o
    const float* R = res + (size_t)bo * 256;
    float* O       = out + (size_t)bo * 65536;
    const int tid  = threadIdx.x;
    const int lane = tid & 31, wave = tid >> 5;
    const int half = lane >> 4, lm = lane & 15;

    const float* casB  = bas;
    const float* casmB = bas + 4096;
    const float* cosB  = bas + 2 * 4096;
    const float* sinB  = bas + 3 * 4096;

    // Phase 1: each wave computes n2-tiles {wave, wave+8} of P and Q (K = 16).
    for (int t = 0; t < 2; ++t) {
        const int nt = (wave + t * 8) * 16;
        v8f accP = {}, accQ = {};
        for (int k = 0; k < 16; k += 4) {
            const int kk = k + 2 * half;
            v2f a;  a.x  = R[lm * 16 + kk];             a.y  = R[lm * 16 + kk + 1];
            v2f bp; bp.x = casB[kk * 256 + nt + lm];    bp.y = casB[(kk + 1) * 256 + nt + lm];
            v2f bq; bq.x = casmB[kk * 256 + nt + lm];   bq.y = casmB[(kk + 1) * 256 + nt + lm];
            accP = wmma_f32(a, bp, accP);
            accQ = wmma_f32(a, bq, accQ);
        }
        for (int r = 0; r < 8; ++r) {
            P[(r + 8 * half) * 256 + nt + lm] = accP[r];
            Q[(r + 8 * half) * 256 + nt + lm] = accQ[r];
        }
    }
    __syncthreads();

    // Phase 2: 16x16 grid of output tiles; wave w emits tiles w, w+8, ...
    for (int tile = wave; tile < 256; tile += 8) {
        const int mt = (tile >> 4) * 16;   // n1 base
        const int nt = (tile & 15) * 16;   // n2 base
        v8f acc = {};
        for (int k = 0; k < 16; k += 4) {
            const int kk = k + 2 * half;
            v2f ac; ac.x = cosB[kk * 256 + mt + lm];  ac.y = cosB[(kk + 1) * 256 + mt + lm];
            v2f as; as.x = sinB[kk * 256 + mt + lm];  as.y = sinB[(kk + 1) * 256 + mt + lm];
            v2f bp; bp.x = P[kk * 256 + nt + lm];     bp.y = P[(kk + 1) * 256 + nt + lm];
            v2f bq; bq.x = Q[kk * 256 + nt + lm];     bq.y = Q[(kk + 1) * 256 + nt + lm];
            acc = wmma_f32(ac, bp, acc);
            acc = wmma_f32(as, bq, acc);
        }
        for (int r = 0; r < 8; ++r)
            O[(size_t)(mt + r + 8 * half) * 256 + nt + lm] = acc[r] * (1.0f / 256.0f);
    }
}

// ---------------------------------------------------------------------------
extern "C" void kernel_launch(void* const* d_in, const int* in_sizes, int n_in,
                              void* d_out, int out_size, void* d_ws, size_t ws_size,
                              hipStream_t stream) {
    const float* x  = (const float*)d_in[0];   // [16,32,256,256]
    const float* w1 = (const float*)d_in[1];   // [32,32,16,16]
    float* out = (float*)d_out;                // [16,32,256,256]

    float* ws  = (float*)d_ws;
    float* bas = ws;                           // 4 * 16*256 floats (64 KB)
    float* Xh  = ws + 4 * 4096;                // 512 * 256 floats (512 KB)
    float* res = ws + 4 * 4096 + 512 * 256;    // 512 * 256 floats (512 KB)

    basis_kernel<<<16, 256, 0, stream>>>(bas);
    dht_fwd_kernel<<<512, 256, 0, stream>>>(x, bas, Xh);
    mode_mix_kernel<<<256, 32, 0, stream>>>(Xh, w1, res);
    dht_inv_kernel<<<512, 256, 0, stream>>>(res, bas, out);
}